// Robust2DUnwrap_27685359190202
// MI455X (gfx1250) — compile-verified
//
#include <hip/hip_runtime.h>
#include <hip/hip_bf16.h>
#include <math.h>

// Robust2DUnwrap on MI455X (gfx1250):
// Replace the dense 12544x12544 solve with a matrix-space Chebyshev iteration
// using A·X = L_M X + X L_N + mu X, driven by V_WMMA_F32_16X16X4_F32.
// Entire state lives in the 320KB WGP LDS; one persistent 32-wave workgroup.

#define MM 112              // image is 112x112
#define MNN (MM * MM)       // 12544 elements per array
#define NT 7                // 112 / 16 tiles per dim
#define NTILES (NT * NT)    // 49 output tiles
#define NTHREADS 1024       // 32 waves (wave32)
#define CHEB_ITERS 128      // q^128 ~ 7e-11 for kappa=121 -> fp32-exact

typedef __attribute__((ext_vector_type(2))) float v2f;
typedef __attribute__((ext_vector_type(8))) float v8f;

__device__ __forceinline__ float wrapf(float v) {
    // (v + 1) mod 2 - 1 with floored (Python-style) modulo
    float y = v + 1.0f;
    y = y - 2.0f * floorf(0.5f * y);
    return y - 1.0f;
}

__global__ __launch_bounds__(NTHREADS, 1)
void Robust2DUnwrap_cheb_wmma(const float* __restrict__ x_in,
                              const float* __restrict__ mu_in,
                              const float* __restrict__ DM,
                              const float* __restrict__ DN,
                              float* __restrict__ out,
                              int iters) {
    extern __shared__ float lds[];
    float* LMs = lds;             // L_M = DM^T DM        (12544 f32)
    float* LNs = lds + 1 * MNN;   // L_N = DN^T DN
    float* Xs  = lds + 2 * MNN;   // solution accumulator
    float* Ds  = lds + 3 * MNN;   // Chebyshev direction (also dx temp in prologue)
    float* Rs  = lds + 4 * MNN;   // residual
    float* Ts  = lds + 5 * MNN;   // t = A*d            (also dy temp in prologue)

    const int tid = threadIdx.x;
    const float mu = mu_in[0];

    // ---------------- Prologue: Gram matrices + wrapped finite differences ----
    for (int e = tid; e < MNN; e += NTHREADS) {
        const int i = e / MM, j = e % MM;
        float lm = 0.0f, ln = 0.0f, ady = 0.0f, adx = 0.0f;
        for (int k = 0; k < MM; ++k) {
            lm  += DM[k * MM + i] * DM[k * MM + j];      // (DM^T DM)[i][j]
            ln  += DN[k * MM + i] * DN[k * MM + j];      // (DN^T DN)[i][j]
            ady += DM[i * MM + k] * x_in[k * MM + j];    // (DM @ X)[i][j]
            adx += x_in[i * MM + k] * DN[j * MM + k];    // (X @ DN^T)[i][j]
        }
        LMs[e] = lm;
        LNs[e] = ln;
        Ts[e] = wrapf(ady);   // dy
        Ds[e] = wrapf(adx);   // dx
    }
    __syncthreads();

    // rhs b = DM^T dy + dx DN + mu x   -> Rs (initial residual, x0 = 0)
    for (int e = tid; e < MNN; e += NTHREADS) {
        const int i = e / MM, j = e % MM;
        float acc = mu * x_in[e];
        for (int k = 0; k < MM; ++k) {
            acc += DM[k * MM + i] * Ts[k * MM + j];   // (DM^T dy)[i][j]
            acc += Ds[i * MM + k] * DN[k * MM + j];   // (dx @ DN)[i][j]
        }
        Rs[e] = acc;
    }
    __syncthreads();

    // ---------------- Chebyshev setup: spec(A) in [mu, 12+mu] (Gershgorin) ----
    const float lmin  = mu;
    const float lmax  = 12.0f + mu;
    const float theta = 0.5f * (lmax + lmin);
    const float delta = 0.5f * (lmax - lmin);
    const float sigma1 = theta / delta;
    float rho = 1.0f / sigma1;

    for (int e = tid; e < MNN; e += NTHREADS) {
        const float d0 = Rs[e] * (1.0f / theta);
        Ds[e] = d0;
        Xs[e] = d0;             // x1 = x0 + d0, x0 = 0
    }
    __syncthreads();

    const int wave = tid >> 5;        // wave32: 32 waves in this workgroup
    const int lane = tid & 31;
    const int lm16 = lane & 15;       // 0..15
    const int lh   = lane >> 4;       // 0 or 1

    // ---------------- Main loop ----------------
    for (int it = 0; it < iters; ++it) {
        // t = L_M*d + d*L_N + mu*d   via V_WMMA_F32_16X16X4_F32, 49 tiles / 32 waves.
        for (int tile = wave; tile < NTILES; tile += 32) {   // wave-uniform: EXEC all 1s
            const int ti = tile / NT, tj = tile % NT;
            const int m0 = ti * 16, n0 = tj * 16;
            v8f acc = {};
            for (int kt = 0; kt < NT; ++kt) {
                const int k0 = kt * 16;
#pragma unroll
                for (int s = 0; s < 4; ++s) {
                    const int ks = k0 + 4 * s;
                    const int kb = ks + 2 * lh;
                    // ---- L_M @ d : A = L_M tile (16x4), B = d tile (4x16)
                    v2f a, b;
                    const float* ap = LMs + (m0 + lm16) * MM + kb;  // 8B contiguous
                    a.x = ap[0];
                    a.y = ap[1];
                    b.x = Ds[kb * MM + n0 + lm16];
                    b.y = Ds[(kb + 1) * MM + n0 + lm16];
                    acc = __builtin_amdgcn_wmma_f32_16x16x4_f32(
                        false, a, false, b, (short)0, acc, false, false);
                    // ---- d @ L_N : A = d tile (16x4), B = L_N tile (4x16)
                    v2f a2, b2;
                    const float* dp = Ds + (m0 + lm16) * MM + kb;   // 8B contiguous
                    a2.x = dp[0];
                    a2.y = dp[1];
                    const float* bp = LNs + kb * MM + n0 + lm16;
                    b2.x = bp[0];
                    b2.y = bp[MM];
                    acc = __builtin_amdgcn_wmma_f32_16x16x4_f32(
                        false, a2, false, b2, (short)0, acc, false, false);
                }
            }
            // acc += mu * d, scatter C/D layout: VGPR r <-> (M = r + 8*lh, N = lm16)
#pragma unroll
            for (int r = 0; r < 8; ++r) {
                const int mmi = m0 + r + 8 * lh;
                const int nni = n0 + lm16;
                Ts[mmi * MM + nni] = acc[r] + mu * Ds[mmi * MM + nni];
            }
        }
        __syncthreads();

        // Chebyshev three-term update (no reductions; rho tracked redundantly per lane)
        const float rho_new = 1.0f / (2.0f * sigma1 - rho);
        const float c1 = rho_new * rho;
        const float c2 = 2.0f * rho_new / delta;
        for (int e = tid; e < MNN; e += NTHREADS) {
            const float rr = Rs[e] - Ts[e];
            Rs[e] = rr;
            const float dd = c1 * Ds[e] + c2 * rr;
            Ds[e] = dd;
            Xs[e] += dd;
        }
        rho = rho_new;
        __syncthreads();
    }

    // ---------------- Epilogue: write solution (row-major [m][n] == reference) ----
    for (int e = tid; e < MNN; e += NTHREADS) {
        out[e] = Xs[e];
    }
}

extern "C" void kernel_launch(void* const* d_in, const int* in_sizes, int n_in,
                              void* d_out, int out_size, void* d_ws, size_t ws_size,
                              hipStream_t stream) {
    // Inputs (setup_inputs order): x (1,1,112,112) f32 | mu (1,) f32 |
    //   A (12544,12544) f32 (UNUSED: Kronecker structure exploited) |
    //   DM (112,112) f32 | DN (112,112) f32
    const float* x  = (const float*)d_in[0];
    const float* mu = (const float*)d_in[1];
    const float* DM = (const float*)d_in[3];
    const float* DN = (const float*)d_in[4];

    const size_t lds_bytes = (size_t)6 * MNN * sizeof(float);  // 301,056 B < 320 KB WGP LDS
    Robust2DUnwrap_cheb_wmma<<<dim3(1), dim3(NTHREADS), lds_bytes, stream>>>(
        x, mu, DM, DN, (float*)d_out, CHEB_ITERS);
}